// GroupedEmbedding_39101382263503
// MI455X (gfx1250) — compile-verified
//
#include <hip/hip_runtime.h>

// GroupedEmbedding gather: out[t*L + l, :] = weights[t, idx[t, l], :]
// T=8 tables, R=100000 rows, D=128 floats, L=262144 lookups/table.
// Pure data movement: 1 GiB gather-read + 1 GiB streaming write, 0 FLOPs.
// HBM roofline @23.3 TB/s => ~92 us floor; memory-bound, WMMA not applicable.
// Strategy: 1 wave32 per row (32 lanes x 16 B = 512 B row), 4 rows per wave
// for in-flight load ILP, wave-uniform index loads forced scalar (SMEM) via
// readfirstlane so gather loads take the saddr-base + lane-voffset form, and
// non-temporal 128-bit stores so the streaming output doesn't evict the
// weight working set (~410 MB, ~47% L2-resident) from the 192 MB L2.

#define TBL 8
#define RPT 100000
#define DIM 128
#define LPK_LOG2 18              // L = 262144 = 2^18
#define NROWS (TBL << LPK_LOG2)  // 2^21 rows total
#define ROWS_PER_WAVE 4

// Native clang ext-vector: accepted by __builtin_nontemporal_store and
// lowers to b128 vector memory ops.
typedef float v4f __attribute__((ext_vector_type(4)));

__global__ __launch_bounds__(256) void grouped_embedding_gather(
    const float* __restrict__ weights,   // [T * R * D]
    const int*   __restrict__ indices,   // [T * L]
    float*       __restrict__ out)       // [T * L * D]
{
    const unsigned lane = threadIdx.x & 31u;
    const unsigned wave = (blockIdx.x * blockDim.x + threadIdx.x) >> 5;
    const unsigned row0 = wave * ROWS_PER_WAVE;

    v4f v[ROWS_PER_WAVE];

    // Phase 1: issue all gather loads (4 x global_load_b128 in flight).
    #pragma unroll
    for (int r = 0; r < ROWS_PER_WAVE; ++r) {
        const unsigned row = row0 + r;
        const unsigned t   = row >> LPK_LOG2;          // table id (uniform)
        // Row index is wave-uniform: force it into an SGPR so the gather
        // load uses the ideal saddr-base + lane-voffset (GVS) form.
        const int i = __builtin_amdgcn_readfirstlane(indices[row]);
        const v4f* src = (const v4f*)(
            weights + ((size_t)t * RPT + (size_t)(unsigned)i) * DIM);
        v[r] = src[lane];
    }

    // Phase 2: streaming non-temporal stores (protect L2 for the tables).
    #pragma unroll
    for (int r = 0; r < ROWS_PER_WAVE; ++r) {
        const unsigned row = row0 + r;
        v4f* dst = (v4f*)(out + (size_t)row * DIM);
        __builtin_nontemporal_store(v[r], dst + lane);
    }
}

extern "C" void kernel_launch(void* const* d_in, const int* in_sizes, int n_in,
                              void* d_out, int out_size, void* d_ws, size_t ws_size,
                              hipStream_t stream) {
    const float* weights = (const float*)d_in[0];   // [T, R, D] float32
    const int*   indices = (const int*)d_in[1];     // [T, L] integer
    float*       out     = (float*)d_out;           // [T*L, D] float32

    // 2^21 rows / 4 rows-per-wave = 2^19 waves; 8 wave32 per 256-thread block
    // -> 65536 blocks. Exact division, no tail.
    const int waves  = NROWS / ROWS_PER_WAVE;
    const int blocks = waves / 8;
    grouped_embedding_gather<<<blocks, 256, 0, stream>>>(weights, indices, out);
}